// StackGCN_37941741093198
// MI455X (gfx1250) — compile-verified
//
#include <hip/hip_runtime.h>

// Problem constants (match reference)
#define N_U    100000
#define N_V    100000
#define D_IN   256
#define D_OUT  320
#define S_REL  5
#define E_NNZ  500000
#define CHUNK  64          // D_OUT / S_REL

typedef __attribute__((ext_vector_type(2))) float v2f;
typedef __attribute__((ext_vector_type(8))) float v8f;

// ---------------------------------------------------------------------------
// GEMM: tmp = X @ W using V_WMMA_F32_16X16X4_F32.
// One wave -> 16 rows x 64 cols output tile. Block = 128 threads (4 waves,
// each wave owns its own 16-row tile). blockIdx.y selects the 64-wide column
// slab (5 slabs = 5 relation chunks), blockIdx.z selects x_u vs x_v.
//
// The W slab (256x64 f32 = 64KB) is staged in LDS once per block in a
// PAIR-INTERLEAVED layout [k/2][n][k&1] so a B fragment (K,K+1 rows at one
// column) is a single 8-byte-aligned ds_load_b64 straight into an even VGPR
// pair — no shuffle movs in the hot loop, conflict-free banking.
// ---------------------------------------------------------------------------
__global__ __launch_bounds__(128) void gemm_wmma_f32(
    const float* __restrict__ x_u, const float* __restrict__ x_v,
    const float* __restrict__ W,  float* __restrict__ tmp_ws)
{
    __shared__ float ldsW[D_IN * CHUNK];          // 64 KB
    const int c0 = blockIdx.y * CHUNK;
    const float* X = (blockIdx.z == 0) ? x_u : x_v;
    float* tmp = tmp_ws + (size_t)blockIdx.z * (size_t)N_U * D_OUT;

    // Stage W[:, c0:c0+64] -> LDS, pair-interleaved: ldsW[(k>>1)*128 + 2n + (k&1)]
    for (int idx = threadIdx.x; idx < D_IN * CHUNK; idx += 128) {
        int k = idx >> 6, n = idx & 63;
        ldsW[(k >> 1) * 128 + 2 * n + (k & 1)] = W[k * D_OUT + c0 + n];
    }
    __syncthreads();

    const int wave = threadIdx.x >> 5;
    const int lane = threadIdx.x & 31;
    const int tile = blockIdx.x * 4 + wave;       // 16-row tile index
    if (tile < (N_U / 16)) {                      // wave-uniform: EXEC stays full
        const int r0    = tile * 16;
        const int nl    = lane & 15;
        const int khalf = lane >> 4;              // 0: K0/K1 half, 1: K2/K3 half
        const int koff  = khalf * 2;

        v8f acc0 = {}, acc1 = {}, acc2 = {}, acc3 = {};
        const float* xrow = X + (size_t)(r0 + nl) * D_IN;
        const v2f*   ldsP = (const v2f*)ldsW;     // pair index = (k>>1)*64 + n

        #pragma unroll 4
        for (int k = 0; k < D_IN; k += 4) {
            // A 16x4 fragment: lane half selects (K0,K1) or (K2,K3) — one b64 load
            v2f a = *(const v2f*)(xrow + k + koff);
            // B 4x16 fragments: one ds_load_b64 each, 4 column sub-tiles
            const v2f* wrow = ldsP + ((k + koff) >> 1) * 64;
            v2f b0 = wrow[nl];
            v2f b1 = wrow[16 + nl];
            v2f b2 = wrow[32 + nl];
            v2f b3 = wrow[48 + nl];
            acc0 = __builtin_amdgcn_wmma_f32_16x16x4_f32(false, a, false, b0, (short)0, acc0, false, false);
            acc1 = __builtin_amdgcn_wmma_f32_16x16x4_f32(false, a, false, b1, (short)0, acc1, false, false);
            acc2 = __builtin_amdgcn_wmma_f32_16x16x4_f32(false, a, false, b2, (short)0, acc2, false, false);
            acc3 = __builtin_amdgcn_wmma_f32_16x16x4_f32(false, a, false, b3, (short)0, acc3, false, false);
        }

        // C/D layout: VGPR r holds M = r + 8*khalf, N = lane&15 (per 16-col tile)
        const int mrow = r0 + 8 * khalf;
        #pragma unroll
        for (int r = 0; r < 8; ++r) {
            float* dst = tmp + (size_t)(mrow + r) * D_OUT + c0 + nl;
            dst[0]  = acc0[r];
            dst[16] = acc1[r];
            dst[32] = acc2[r];
            dst[48] = acc3[r];
        }
    }
}

// ---------------------------------------------------------------------------
// Zero the output (segment_sum accumulates into it).
// ---------------------------------------------------------------------------
__global__ __launch_bounds__(256) void zero_f4(float4* __restrict__ p)
{
    size_t i = (size_t)blockIdx.x * blockDim.x + threadIdx.x;
    p[i] = make_float4(0.f, 0.f, 0.f, 0.f);
}

// ---------------------------------------------------------------------------
// Fused bidirectional COO SpMM scatter. One wave per edge:
//   z_u[row, s*64:+64] += val * tmp_v[col, s*64:+64]
//   z_v[col, s*64:+64] += val * tmp_u[row, s*64:+64]
// Per-lane float2 gather (global_load_b64, 256B contiguous per wave) then
// two global_atomic_add_f32. blockIdx.y = relation s.
// ---------------------------------------------------------------------------
__global__ __launch_bounds__(256) void spmm_scatter(
    const float* __restrict__ tmp_ws, float* __restrict__ z_out,
    const int* __restrict__ sup_rows, const int* __restrict__ sup_cols,
    const float* __restrict__ sup_vals)
{
    const int s    = blockIdx.y;
    const int lane = threadIdx.x & 31;
    const int wave = threadIdx.x >> 5;
    const long e   = (long)blockIdx.x * 8 + wave;
    if (e >= E_NNZ) return;                       // wave-uniform

    const long ei = (long)s * E_NNZ + e;
    const int   r = sup_rows[ei];
    const int   c = sup_cols[ei];
    const float v = sup_vals[ei];

    const float* tmp_u = tmp_ws;
    const float* tmp_v = tmp_ws + (size_t)N_U * D_OUT;
    float*       z_u   = z_out;
    float*       z_v   = z_out  + (size_t)N_U * D_OUT;

    const int co = s * CHUNK;
    const int l2 = 2 * lane;
    const float* su = tmp_u + (size_t)r * D_OUT + co;
    const float* sv = tmp_v + (size_t)c * D_OUT + co;
    float*       du = z_u   + (size_t)r * D_OUT + co;
    float*       dv = z_v   + (size_t)c * D_OUT + co;

    float2 av = *(const float2*)(sv + l2);
    atomicAdd(&du[l2],     v * av.x);
    atomicAdd(&du[l2 + 1], v * av.y);

    float2 au = *(const float2*)(su + l2);
    atomicAdd(&dv[l2],     v * au.x);
    atomicAdd(&dv[l2 + 1], v * au.y);
}

// ---------------------------------------------------------------------------
// In-place ReLU over the concatenated [z_u; z_v] output.
// ---------------------------------------------------------------------------
__global__ __launch_bounds__(256) void relu_f4(float4* __restrict__ p)
{
    size_t i = (size_t)blockIdx.x * blockDim.x + threadIdx.x;
    float4 t = p[i];
    t.x = fmaxf(t.x, 0.f);
    t.y = fmaxf(t.y, 0.f);
    t.z = fmaxf(t.z, 0.f);
    t.w = fmaxf(t.w, 0.f);
    p[i] = t;
}

// ---------------------------------------------------------------------------
extern "C" void kernel_launch(void* const* d_in, const int* in_sizes, int n_in,
                              void* d_out, int out_size, void* d_ws, size_t ws_size,
                              hipStream_t stream)
{
    const float* x_u      = (const float*)d_in[0];
    const float* x_v      = (const float*)d_in[1];
    const float* W        = (const float*)d_in[2];
    const float* sup_vals = (const float*)d_in[3];
    const int*   sup_rows = (const int*)d_in[4];
    const int*   sup_cols = (const int*)d_in[5];

    float* tmp = (float*)d_ws;   // needs 2 * 100000 * 320 * 4 = 256 MB scratch

    // 1) tmp_u / tmp_v = X @ W  (WMMA f32)
    dim3 gGemm((N_U / 16 + 3) / 4, D_OUT / CHUNK, 2);   // 1563 x 5 x 2
    gemm_wmma_f32<<<gGemm, 128, 0, stream>>>(x_u, x_v, W, tmp);

    // 2) zero outputs (64M floats = 16M float4 = 62500 * 256)
    zero_f4<<<62500, 256, 0, stream>>>((float4*)d_out);

    // 3) bidirectional SpMM scatter (one wave per edge, 8 edges/block)
    dim3 gSp(E_NNZ / 8, S_REL, 1);                      // 62500 x 5
    spmm_scatter<<<gSp, 256, 0, stream>>>(tmp, (float*)d_out,
                                          sup_rows, sup_cols, sup_vals);

    // 4) ReLU in place
    relu_f4<<<62500, 256, 0, stream>>>((float4*)d_out);
}